// RelationalGATLayer_33603824124032
// MI455X (gfx1250) — compile-verified
//
#include <hip/hip_runtime.h>

#define N_NODES 50000
#define N_EDGES 1600000
#define IN_F 256
#define OUT_F 128
#define N_REL 500
#define NEG_SLOPE 0.2f
#define GAT_EPS 1e-10f

typedef float v2f __attribute__((ext_vector_type(2)));
typedef float v8f __attribute__((ext_vector_type(8)));

// ---------------------------------------------------------------------------
// Kernel 1: Wh = h @ W  via V_WMMA_F32_16X16X4_F32 (fp32 WMMA, exact math).
// One wave computes one 16x16 tile of Wh. 8 waves/block cover OUT_F=128.
// A layout (32-bit A 16x4): lanes 0-15 -> M=lane, K={0,1}; lanes 16-31 -> K={2,3}.
// B layout (4x16):          VGPR0 holds K=0 (lanes 0-15) / K=2 (lanes 16-31),
//                           VGPR1 holds K=1 / K=3, N = lane&15.
// C/D layout: VGPR i -> M=i (lanes 0-15), M=i+8 (lanes 16-31), N = lane&15.
// ---------------------------------------------------------------------------
__global__ __launch_bounds__(256)
void wh_gemm_kernel(const float* __restrict__ h, const float* __restrict__ W,
                    float* __restrict__ Wh) {
  const int lane = threadIdx.x & 31;
  const int wv   = threadIdx.x >> 5;        // N-tile within block (0..7)
  const int tm   = blockIdx.x * 16;         // M tile base (3125 blocks)
  const int tn   = wv * 16;                 // N tile base
  const int r    = lane & 15;
  const int kh   = (lane >> 4) << 1;        // 0 or 2

  const float* ap = h + (size_t)(tm + r) * IN_F + kh;        // contiguous pair
  const float* bp = W + (size_t)kh * OUT_F + (tn + r);       // stride OUT_F pair

  v8f acc = {};
  #pragma unroll 4
  for (int k0 = 0; k0 < IN_F; k0 += 4) {
    v2f a = *(const v2f*)ap;                 // h[tm+r][k0+kh+{0,1}]
    v2f b;
    b.x = bp[0];                             // W[k0+kh+0][tn+r]
    b.y = bp[OUT_F];                         // W[k0+kh+1][tn+r]
    acc = __builtin_amdgcn_wmma_f32_16x16x4_f32(
        /*neg_a=*/false, a, /*neg_b=*/false, b,
        /*c_mod=*/(short)0, acc, /*reuse_a=*/false, /*reuse_b=*/false);
    ap += 4;
    bp += 4 * OUT_F;
  }

  const int n     = tn + r;
  const int mbase = tm + ((lane >> 4) << 3);
  float* op = Wh + (size_t)mbase * OUT_F + n;
  #pragma unroll
  for (int i = 0; i < 8; ++i) op[(size_t)i * OUT_F] = acc[i];
}

// ---------------------------------------------------------------------------
// Kernel 2: per-row dots. rows [0,N_NODES): s_i=Wh.a_i, s_j=Wh.a_j.
//           rows [N_NODES, N_NODES+N_REL): s_rel = rel_emb . a_r.
// One wave per row; lane handles 4 contiguous floats; shuffle reduction.
// ---------------------------------------------------------------------------
__global__ __launch_bounds__(256)
void score_kernel(const float* __restrict__ Wh, const float* __restrict__ rel_emb,
                  const float* __restrict__ a,
                  float* __restrict__ s_i, float* __restrict__ s_j,
                  float* __restrict__ s_rel) {
  const int lane = threadIdx.x & 31;
  const int row  = (int)(((size_t)blockIdx.x * blockDim.x + threadIdx.x) >> 5);
  if (row >= N_NODES + N_REL) return;
  const int c = lane << 2;                  // 32 lanes * 4 = 128 cols

  if (row < N_NODES) {
    const float4 w  = *(const float4*)(Wh + (size_t)row * OUT_F + c);
    const float4 ai = *(const float4*)(a + c);
    const float4 aj = *(const float4*)(a + OUT_F + c);
    float pi = w.x * ai.x + w.y * ai.y + w.z * ai.z + w.w * ai.w;
    float pj = w.x * aj.x + w.y * aj.y + w.z * aj.z + w.w * aj.w;
    #pragma unroll
    for (int o = 16; o > 0; o >>= 1) {
      pi += __shfl_xor(pi, o, 32);
      pj += __shfl_xor(pj, o, 32);
    }
    if (lane == 0) { s_i[row] = pi; s_j[row] = pj; }
  } else {
    const int rr = row - N_NODES;
    const float4 w  = *(const float4*)(rel_emb + (size_t)rr * OUT_F + c);
    const float4 ar = *(const float4*)(a + 2 * OUT_F + c);
    float p = w.x * ar.x + w.y * ar.y + w.z * ar.z + w.w * ar.w;
    #pragma unroll
    for (int o = 16; o > 0; o >>= 1) p += __shfl_xor(p, o, 32);
    if (lane == 0) s_rel[rr] = p;
  }
}

// Monotonic order-preserving float<->uint key (for atomic max of floats).
__device__ __forceinline__ unsigned fkey(float f) {
  unsigned u = __float_as_uint(f);
  return (u & 0x80000000u) ? ~u : (u | 0x80000000u);
}
__device__ __forceinline__ float funkey(unsigned k) {
  return (k & 0x80000000u) ? __uint_as_float(k & 0x7fffffffu)
                           : __uint_as_float(~k);
}

// ---------------------------------------------------------------------------
// Kernel 3: e = leaky_relu(s_i[row] + s_j[col] + s_rel[et]); block-reduce max.
// ---------------------------------------------------------------------------
__global__ __launch_bounds__(256)
void edge_score_kernel(const int* __restrict__ rows, const int* __restrict__ cols,
                       const int* __restrict__ et,
                       const float* __restrict__ s_i, const float* __restrict__ s_j,
                       const float* __restrict__ s_rel,
                       float* __restrict__ e_out, unsigned* __restrict__ maxKey) {
  __shared__ unsigned sk[256];
  const int i = blockIdx.x * blockDim.x + threadIdx.x;
  unsigned k = 0u;   // identity (maps below key(-inf))
  if (i < N_EDGES) {
    float e = s_i[rows[i]] + s_j[cols[i]] + s_rel[et[i]];
    e = (e > 0.0f) ? e : NEG_SLOPE * e;
    e_out[i] = e;
    k = fkey(e);
  }
  sk[threadIdx.x] = k;
  __syncthreads();
  #pragma unroll
  for (int s = 128; s > 0; s >>= 1) {
    if (threadIdx.x < s) {
      unsigned o = sk[threadIdx.x + s];
      if (o > sk[threadIdx.x]) sk[threadIdx.x] = o;
    }
    __syncthreads();
  }
  if (threadIdx.x == 0) atomicMax(maxKey, sk[0]);
}

// ---------------------------------------------------------------------------
// Kernel 4: exp_e = exp(e - max); segment-sum into sum_exp via f32 atomics.
// Rewrites e_buf in place with exp_e.
// ---------------------------------------------------------------------------
__global__ __launch_bounds__(256)
void edge_exp_kernel(const int* __restrict__ rows, float* __restrict__ e_buf,
                     const unsigned* __restrict__ maxKey,
                     float* __restrict__ sum_exp) {
  const int i = blockIdx.x * blockDim.x + threadIdx.x;
  if (i >= N_EDGES) return;
  const float m = funkey(*maxKey);
  const float p = expf(e_buf[i] - m);
  e_buf[i] = p;
  atomicAdd(&sum_exp[rows[i]], p);
}

// ---------------------------------------------------------------------------
// Kernel 5: h_prime[row] += alpha * Wh[col]. One wave per edge, lane = 4 cols.
// Wh and h_prime are both L2-resident (25.6 MB each, 192 MB L2).
// ---------------------------------------------------------------------------
__global__ __launch_bounds__(256)
void scatter_kernel(const int* __restrict__ rows, const int* __restrict__ cols,
                    const float* __restrict__ exp_e, const float* __restrict__ sum_exp,
                    const float* __restrict__ Wh, float* __restrict__ hp) {
  const int lane = threadIdx.x & 31;
  const long long edge =
      (((long long)blockIdx.x * blockDim.x) + threadIdx.x) >> 5;
  if (edge >= N_EDGES) return;
  const int r = rows[edge];
  const int c = cols[edge];
  const float alpha = exp_e[edge] / (sum_exp[r] + GAT_EPS);
  const int off = lane << 2;
  const float4 w = *(const float4*)(Wh + (size_t)c * OUT_F + off);
  float* d = hp + (size_t)r * OUT_F + off;
  atomicAdd(d + 0, alpha * w.x);
  atomicAdd(d + 1, alpha * w.y);
  atomicAdd(d + 2, alpha * w.z);
  atomicAdd(d + 3, alpha * w.w);
}

// ---------------------------------------------------------------------------
// Init: zero accumulators (d_out, sum_exp, maxKey) — required every launch.
// ---------------------------------------------------------------------------
__global__ void init_kernel(float* __restrict__ hp, float* __restrict__ sum_exp,
                            unsigned* __restrict__ maxKey) {
  const long long i = (long long)blockIdx.x * blockDim.x + threadIdx.x;
  if (i < (long long)N_NODES * OUT_F) hp[i] = 0.0f;
  if (i < N_NODES) sum_exp[i] = 0.0f;
  if (i == 0) *maxKey = 0u;
}

// ---------------------------------------------------------------------------
// Final: out = elu(h_prime), in place on d_out.
// ---------------------------------------------------------------------------
__global__ void elu_kernel(float* __restrict__ hp) {
  const long long i = (long long)blockIdx.x * blockDim.x + threadIdx.x;
  if (i >= (long long)N_NODES * OUT_F) return;
  const float v = hp[i];
  hp[i] = (v > 0.0f) ? v : (expf(v) - 1.0f);
}

extern "C" void kernel_launch(void* const* d_in, const int* in_sizes, int n_in,
                              void* d_out, int out_size, void* d_ws, size_t ws_size,
                              hipStream_t stream) {
  const float* h       = (const float*)d_in[0];
  const int*   rows    = (const int*)d_in[1];
  const int*   cols    = (const int*)d_in[2];
  const int*   et      = (const int*)d_in[3];
  const float* W       = (const float*)d_in[4];
  const float* rel_emb = (const float*)d_in[5];
  const float* a       = (const float*)d_in[6];
  float*       out     = (float*)d_out;

  // Workspace carve-up (floats): Wh | e/exp | s_i | s_j | sum_exp | s_rel | maxKey
  float* ws = (float*)d_ws;
  size_t o = 0;
  float* Wh      = ws + o; o += (size_t)N_NODES * OUT_F;   // 6.40M
  float* ebuf    = ws + o; o += (size_t)N_EDGES;           // 1.60M
  float* s_i     = ws + o; o += (size_t)N_NODES;
  float* s_j     = ws + o; o += (size_t)N_NODES;
  float* sum_exp = ws + o; o += (size_t)N_NODES;
  float* s_rel   = ws + o; o += 512;                       // padded
  unsigned* maxKey = (unsigned*)(ws + o);

  const long long outElems = (long long)N_NODES * OUT_F;   // 6.4M

  init_kernel<<<(int)((outElems + 255) / 256), 256, 0, stream>>>(out, sum_exp, maxKey);

  wh_gemm_kernel<<<N_NODES / 16, 256, 0, stream>>>(h, W, Wh);     // 3125 blocks

  score_kernel<<<((N_NODES + N_REL) * 32 + 255) / 256, 256, 0, stream>>>(
      Wh, rel_emb, a, s_i, s_j, s_rel);

  edge_score_kernel<<<(N_EDGES + 255) / 256, 256, 0, stream>>>(
      rows, cols, et, s_i, s_j, s_rel, ebuf, maxKey);

  edge_exp_kernel<<<(N_EDGES + 255) / 256, 256, 0, stream>>>(
      rows, ebuf, maxKey, sum_exp);

  scatter_kernel<<<N_EDGES / 8, 256, 0, stream>>>(                 // 200000 blocks
      rows, cols, ebuf, sum_exp, Wh, out);

  elu_kernel<<<(int)((outElems + 255) / 256), 256, 0, stream>>>(out);
}